// FastFoodLayerDense_26139170964011
// MI455X (gfx1250) — compile-verified
//
#include <hip/hip_runtime.h>

typedef float v2f __attribute__((ext_vector_type(2)));
typedef float v8f __attribute__((ext_vector_type(8)));

#define NROWS 16
#define DDIM  1024
#define TPB   256

// Bank-conflict-avoiding XOR swizzle: physical LDS index for (row, col).
// XORs column bits 2..5 with the row index -> 16 distinct bank groups for
// column-parallel accesses across rows.  FWHT butterflies and per-element
// WMMA operand loads are both invariant under this per-row column XOR.
__device__ __forceinline__ int lidx(int r, int c) {
    return (r << 10) + (c ^ ((r << 2) & 0x3C));
}

// Sylvester-Hadamard entry: H[i][j] = (-1)^popcount(i&j)
__device__ __forceinline__ float hsign(int i, int j) {
    return (__builtin_popcount(i & j) & 1) ? -1.0f : 1.0f;
}

// In-place FWHT-1024 on 16 rows held in LDS (swizzled addressing).
// H1024 = H64 (x) H16 with index = a*16 + b:
//   Stage A: Y(16x16) = X(16x16) * H16 per 16-block, via chained f32 WMMA
//   Stage B: 6 add/sub butterfly stages across the 64 blocks (H64)
__device__ void fwht_tile(float* buf, int t) {
    const int lane  = t & 31;
    const int wave  = t >> 5;          // 0..7
    const int m     = lane & 15;       // A row / B,D column
    const int khalf = (lane >> 4) * 2; // K sub-offset for this lane half

    // B-matrix chunks (4x16 each): B_c[k][n] = H16[c*4+k][n], built in VALU
    v2f bmat[4];
#pragma unroll
    for (int c = 0; c < 4; ++c) {
        bmat[c].x = hsign(c * 4 + khalf + 0, m);
        bmat[c].y = hsign(c * 4 + khalf + 1, m);
    }

    // ---- Stage A: 8 blocks per wave, K=16 via 4 chained 16x16x4 WMMAs ----
#pragma unroll
    for (int it = 0; it < 8; ++it) {
        const int blk   = wave + it * 8;
        const int cbase = blk << 4;
        v8f acc = {0.f, 0.f, 0.f, 0.f, 0.f, 0.f, 0.f, 0.f};
#pragma unroll
        for (int c = 0; c < 4; ++c) {
            v2f a;  // A 16x4 f32 layout: lane<16 -> M=lane, K={0,1}; lane>=16 -> K={2,3}
            a.x = buf[lidx(m, cbase + c * 4 + khalf + 0)];
            a.y = buf[lidx(m, cbase + c * 4 + khalf + 1)];
            acc = __builtin_amdgcn_wmma_f32_16x16x4_f32(
                false, a, false, bmat[c], (short)0, acc, false, false);
        }
        // D layout: lane<16 -> rows 0..7 col=lane ; lane>=16 -> rows 8..15 col=lane-16
        const int rbase = (lane >> 4) * 8;
#pragma unroll
        for (int r = 0; r < 8; ++r)
            buf[lidx(rbase + r, cbase + m)] = acc[r];
    }
    __syncthreads();

    // ---- Stage B: butterflies across blocks, strides 1,2,4,8,16,32 ----
#pragma unroll
    for (int st = 0; st < 6; ++st) {
        const int h = 1 << st;                 // stride in 16-element blocks
        for (int p = t; p < 8192; p += TPB) {  // 16 rows * 512 pairs/row
            const int row = p >> 9;
            const int q   = p & 511;
            const int pr  = q >> 4;            // a-pair index 0..31
            const int b   = q & 15;
            const int alo = ((pr & ~(h - 1)) << 1) | (pr & (h - 1));
            const int i = lidx(row, (alo << 4) + b);
            const int j = lidx(row, ((alo + h) << 4) + b);
            const float u = buf[i], v = buf[j];
            buf[i] = u + v;
            buf[j] = u - v;
        }
        __syncthreads();
    }
}

__global__ void __launch_bounds__(TPB)
fastfood_kernel(const float* __restrict__ x,   const float* __restrict__ Bsg,
                const float* __restrict__ G,   const float* __restrict__ Sv,
                const float* __restrict__ bias, const int* __restrict__ P,
                float* __restrict__ out)
{
    __shared__ float buf[NROWS * DDIM];        // 64 KB
    const int t    = threadIdx.x;
    const int row0 = blockIdx.x * NROWS;
    const int s    = blockIdx.y;
    const int sD   = s * DDIM;

    // ---- Phase 0: load x tile (float4), apply sign diag B, store to LDS ----
#pragma unroll
    for (int i = 0; i < 16; ++i) {
        const int f4  = t + TPB * i;           // 4096 float4 = 16 rows x 1024
        const int row = f4 >> 8;
        const int col = (f4 & 255) << 2;
        const float4 xv = *(const float4*)(x + (size_t)(row0 + row) * DDIM + col);
        const float4 bv = *(const float4*)(Bsg + sD + col);
        float4 hv;
        hv.x = xv.x * bv.x; hv.y = xv.y * bv.y;
        hv.z = xv.z * bv.z; hv.w = xv.w * bv.w;
        *(float4*)(buf + lidx(row, col)) = hv; // swizzle preserves 16B alignment
    }
    __syncthreads();

    fwht_tile(buf, t);                         // first Hadamard

    // ---- Phase 2: intra-row permutation gather + diag G (reg staging) ----
#pragma unroll
    for (int half = 0; half < 2; ++half) {     // rows [0,8) then [8,16)
        float tmp[32];
#pragma unroll
        for (int i = 0; i < 32; ++i) {
            const int e   = half * 8192 + t + TPB * i;
            const int row = e >> 10;
            const int d   = e & 1023;
            tmp[i] = buf[lidx(row, P[sD + d] - sD)] * G[sD + d];
        }
        __syncthreads();
#pragma unroll
        for (int i = 0; i < 32; ++i) {
            const int e = half * 8192 + t + TPB * i;
            buf[lidx(e >> 10, e & 1023)] = tmp[i];
        }
        __syncthreads();
    }

    fwht_tile(buf, t);                         // second Hadamard

    // ---- Phase 3: scale by S/(sigma*sqrt(D)), cos/sin, /sqrt(D), + bias ----
    const float inv32 = 0.03125f;              // 1/sqrt(1024)
    for (int i = 0; i < 64; ++i) {
        const int e   = t + TPB * i;
        const int row = e >> 10;
        const int d   = e & 1023;
        const float z = buf[lidx(row, d)] * Sv[sD + d] * inv32;
        float sn, cs;
        __sincosf(z, &sn, &cs);
        const size_t ob = (size_t)(row0 + row) * 8192;
        out[ob + sD + d]        = inv32 * cs + bias[sD + d];
        out[ob + 4096 + sD + d] = inv32 * sn + bias[4096 + sD + d];
    }
}

extern "C" void kernel_launch(void* const* d_in, const int* in_sizes, int n_in,
                              void* d_out, int out_size, void* d_ws, size_t ws_size,
                              hipStream_t stream) {
    (void)in_sizes; (void)n_in; (void)out_size; (void)d_ws; (void)ws_size;
    const float* x    = (const float*)d_in[0];
    const float* Bsg  = (const float*)d_in[1];
    const float* G    = (const float*)d_in[2];
    const float* Sv   = (const float*)d_in[3];
    const float* bias = (const float*)d_in[4];
    // d_in[5] is the dense Hadamard matrix H: unused, we apply H implicitly
    const int*   P    = (const int*)d_in[6];
    float* out = (float*)d_out;

    dim3 grid(8192 / NROWS, 4);   // 512 row-tiles x 4 stacks
    hipLaunchKernelGGL(fastfood_kernel, grid, dim3(TPB), 0, stream,
                       x, Bsg, G, Sv, bias, P, out);
}